// CTCLayer_47227460386998
// MI455X (gfx1250) — compile-verified
//
#include <hip/hip_runtime.h>

// CTC forward loss (K.ctc_batch_cost) for B=512, T=512, C=128, L=64.
//
// Wave32-resident scan: one wave per batch element, each lane owns CH=5
// consecutive extended-label states in registers. Per time step the only
// cross-lane traffic is two __shfl_up(1) for the previous lane's boundary
// alphas; the 3-way logsumexp is pure VALU (v_max3/v_exp/v_log). No barriers.
//
// Data path (CDNA5 async): each wave ring-buffers R=8 rows (512 B each) of its
// y_pred slab in LDS via GLOBAL_LOAD_ASYNC_TO_LDS_B128 (one instruction copies
// a whole row: 32 lanes x 16 B), synchronized with s_wait_asynccnt (async
// loads complete in order per the ISA). The time loop is unrolled in blocks of
// R so ring slots are compile-time constants: gathers become ds_load_b32 off
// loop-invariant address VGPRs with immediate slot offsets, and refill
// addresses fold into the 24-bit global IOFFSET.

#define NEGV (-1.0e9f)
#define EPSF (1e-7f)

constexpr int T   = 512;
constexpr int C   = 128;
constexpr int L   = 64;
constexpr int S   = 2 * L + 1;   // 129 extended states
constexpr int CH  = 5;           // states per lane (32*5 = 160 >= 129, padded)
constexpr int WPB = 8;           // waves per block
constexpr int R   = 8;           // LDS ring depth (rows of y_pred per wave)
constexpr int NB  = T / R;       // 64 blocks of R time steps

#if defined(__HIP_DEVICE_COMPILE__) &&                                         \
    __has_builtin(__builtin_amdgcn_global_load_async_to_lds_b128) &&           \
    __has_builtin(__builtin_amdgcn_s_wait_asynccnt)
#define USE_ASYNC 1
typedef int v4i __attribute__((vector_size(16)));
typedef __attribute__((address_space(1))) v4i* g_v4i_ptr;   // global (AS1)
typedef __attribute__((address_space(3))) v4i* l_v4i_ptr;   // LDS (AS3)
#else
#define USE_ASYNC 0
#endif

// one alpha recursion step; p holds this step's gathered probabilities
__device__ __forceinline__ void alpha_update(float (&a)[CH], const float (&p)[CH],
                                             const bool (&skip)[CH], int lane)
{
    float lp[CH];
#pragma unroll
    for (int i = 0; i < CH; ++i) lp[i] = __logf(p[i] + EPSF);

    float pm1 = __shfl_up(a[CH - 1], 1, 32);   // old alpha at state base-1
    float pm2 = __shfl_up(a[CH - 2], 1, 32);   // old alpha at state base-2
    if (lane == 0) { pm1 = NEGV; pm2 = NEGV; }

    float na[CH];
#pragma unroll
    for (int i = 0; i < CH; ++i) {
        const float a1 = a[i];                                   // stay
        const float a2 = (i >= 1) ? a[i - 1] : pm1;              // advance 1
        float       a3 = (i >= 2) ? a[i - 2]                     // advance 2
                                  : ((i == 1) ? pm1 : pm2);
        a3 = skip[i] ? a3 : NEGV;
        const float m  = fmaxf(fmaxf(a1, a2), a3);               // v_max3_f32
        const float sm = __expf(a1 - m) + __expf(a2 - m) + __expf(a3 - m);
        na[i] = lp[i] + m + __logf(sm);
    }
#pragma unroll
    for (int i = 0; i < CH; ++i) a[i] = na[i];
}

__global__ __launch_bounds__(32 * WPB)
void ctc_scan_kernel(const int* __restrict__ y_true,
                     const float* __restrict__ y_pred,
                     float* __restrict__ out, int B)
{
#if USE_ASYNC
    __shared__ float ring[WPB][R][C];   // 8*8*128*4 = 32 KB per workgroup
#endif
    const int lane = threadIdx.x & 31;
    const int wave = threadIdx.x >> 5;
    const int b    = blockIdx.x * WPB + wave;
    if (b >= B) return;                 // wave-uniform

    const float* __restrict__ yb  = y_pred + (size_t)b * T * C;
    const int*   __restrict__ lab = y_true + (size_t)b * L;

    // ---- per-lane static lattice info (ext symbol + skip-transition flag) ----
    int  ext[CH];
    bool skip[CH];
#pragma unroll
    for (int i = 0; i < CH; ++i) {
        const int  s   = lane * CH + i;
        const bool odd = (s & 1) != 0;
        const int  li  = (s - 1) >> 1;                  // label index when odd
        ext[i]  = (odd && s < S) ? lab[li] : (C - 1);   // even/padded -> blank
        skip[i] = (odd && s >= 3 && s < S) ? (lab[li] != lab[li - 1]) : false;
    }

    float a[CH];

#if USE_ASYNC
    float* const myring = &ring[wave][0][0];

    // async-copy one 512B row (row of yb) into ring slot 'slot'
    auto refill = [&](int row, int slot) {
        __builtin_amdgcn_global_load_async_to_lds_b128(
            (g_v4i_ptr)(yb + (size_t)row * C + lane * 4),
            (l_v4i_ptr)(myring + slot * C + lane * 4), 0, 0);
    };
    // per-lane 5-point gather from ring slot 'slot' (slot is a constant ->
    // ds_load_b32 with immediate offset off loop-invariant address VGPRs)
    auto gather = [&](float (&p)[CH], int slot) {
#pragma unroll
        for (int i = 0; i < CH; ++i) p[i] = myring[slot * C + ext[i]];
    };

    // ---- prefill the ring with rows 0..R-1 (ASYNCcnt = R) ----
#pragma unroll
    for (int r = 0; r < R; ++r) refill(r, r);

    // ---- block 0: t = 0 (alpha init) then t = 1..R-1 ----
    {
        __builtin_amdgcn_s_wait_asynccnt(R - 1);        // row 0 landed (in-order)
        float p[CH];
        gather(p, 0);
#pragma unroll
        for (int i = 0; i < CH; ++i) {
            const int s = lane * CH + i;
            a[i] = (s <= 1) ? __logf(p[i] + EPSF) : NEGV;
        }
        refill(R, 0);                                   // row 8 -> slot 0
#pragma unroll
        for (int tt = 1; tt < R; ++tt) {
            __builtin_amdgcn_s_wait_asynccnt(R - 1);    // row tt landed
            float pt[CH];
            gather(pt, tt);
            alpha_update(a, pt, skip, lane);
            refill(tt + R, tt);                         // rows 9..15
        }
    }

    // ---- steady state: blocks 1..NB-2 (t = 8..503), unconditional refills ----
    for (int tb = 1; tb < NB - 1; ++tb) {
        const int t0 = tb * R;
#pragma unroll
        for (int tt = 0; tt < R; ++tt) {
            __builtin_amdgcn_s_wait_asynccnt(R - 1);    // row t0+tt landed
            float p[CH];
            gather(p, tt);
            alpha_update(a, p, skip, lane);
            refill(t0 + tt + R, tt);                    // rows 16..511
        }
    }

    // ---- final block: t = 504..511, ring drains ----
#pragma unroll
    for (int tt = 0; tt < R; ++tt) {
        __builtin_amdgcn_s_wait_asynccnt(R - 1);
        float p[CH];
        gather(p, tt);
        alpha_update(a, p, skip, lane);
    }
#else
    // ---- fallback: direct global gathers + deep prefetch ----
    {
        float p[CH];
#pragma unroll
        for (int i = 0; i < CH; ++i) p[i] = yb[ext[i]];
#pragma unroll
        for (int i = 0; i < CH; ++i) {
            const int s = lane * CH + i;
            a[i] = (s <= 1) ? __logf(p[i] + EPSF) : NEGV;
        }
    }
    for (int t = 1; t < T; ++t) {
        int tp = t + R; tp = (tp < T) ? tp : (T - 1);
        __builtin_prefetch(yb + (size_t)tp * C + lane * 4, 0, 3);
        const float* __restrict__ row = yb + (size_t)t * C;
        float p[CH];
#pragma unroll
        for (int i = 0; i < CH; ++i) p[i] = row[ext[i]];
        alpha_update(a, p, skip, lane);
    }
#endif

    // ---- final: -logsumexp(alpha[127], alpha[128]); both live in lane 25
    //      (25*5+2 = 127, 25*5+3 = 128) ----
    if (lane == 25) {
        const float x = a[2], y = a[3];
        const float m = fmaxf(x, y);
        const float ll = m + __logf(__expf(x - m) + __expf(y - m));
        out[b] = -ll;
    }
}

extern "C" void kernel_launch(void* const* d_in, const int* in_sizes, int n_in,
                              void* d_out, int out_size, void* d_ws, size_t ws_size,
                              hipStream_t stream) {
    const int*   y_true = (const int*)d_in[0];    // [B, L] int32
    const float* y_pred = (const float*)d_in[1];  // [B, T, C] float32 probs
    float*       out    = (float*)d_out;          // [B, 1] float32
    const int B = out_size;                       // 512

    dim3 block(32 * WPB);
    dim3 grid((B + WPB - 1) / WPB);
    ctc_scan_kernel<<<grid, block, 0, stream>>>(y_true, y_pred, out, B);
}